// BEVDetectionLoss_61125974556946
// MI455X (gfx1250) — compile-verified
//
#include <hip/hip_runtime.h>
#include <hip/hip_bf16.h>

// ---------------- BEV constants (match reference) ----------------
#define X_MIN   (-51.2f)
#define Y_MIN   (-51.2f)
#define X_MAX   ( 51.2f)
#define Y_MAX   ( 51.2f)
#define RES     ( 0.2f)
#define BEV_W   512
#define BEV_H   512
#define NUM_CELLS (BEV_W * BEV_H)   // 262144

#define K1_BLOCKS  1024
#define K1_THREADS 256

typedef __attribute__((ext_vector_type(2))) float v2f;
typedef __attribute__((ext_vector_type(8))) float v8f;

// Numerically stable softplus, matches jax.nn.softplus = log1p(exp(x)) stable form.
__device__ __forceinline__ float softplusf(float x) {
    float ax = fabsf(x);
    return fmaxf(x, 0.0f) + log1pf(expf(-ax));
}

// Exact f32 sum across all 32 lanes of a wave using V_WMMA_F32_16X16X4_F32.
// A-matrix 16x4 layout: lanes 0-15 supply K=0 (VGPR0)/K=1 (VGPR1);
// lanes 16-31 supply K=2/K=3.  Put v in VGPR0, 0 in VGPR1, B = ones:
//   D[m][n] = v[m] + v[m+16]  (C/D layout: lane l<16 VGPR r = D[r][l],
//                              lane l>=16 VGPR r = D[r+8][l-16])
// Sum of a lane's 8 D VGPRs = half-sum; one xor-16 shuffle finishes it.
// Requires EXEC == all ones (call with all lanes active, no divergence).
__device__ __forceinline__ float wave_sum_wmma(float v) {
    v2f a; a.x = v;     a.y = 0.0f;
    v2f b; b.x = 1.0f;  b.y = 1.0f;
    v8f c = {};
    c = __builtin_amdgcn_wmma_f32_16x16x4_f32(
            /*neg_a=*/false, a, /*neg_b=*/false, b,
            /*c_mod=*/(short)0, c, /*reuse_a=*/false, /*reuse_b=*/false);
    float h = ((c[0] + c[1]) + (c[2] + c[3])) + ((c[4] + c[5]) + (c[6] + c[7]));
    return h + __shfl_xor(h, 16, 32);
}

// ---------------- Kernel 1: streaming softplus sum over obj_logits ----------
// Deterministic: each block writes one partial (fixed reduction order).
__global__ void __launch_bounds__(K1_THREADS)
bev_obj_softplus_partials(const float* __restrict__ obj,
                          float* __restrict__ partials, int n) {
    const int tid    = blockIdx.x * blockDim.x + threadIdx.x;
    const int stride = gridDim.x * blockDim.x;

    float acc = 0.0f;
    const int n4 = n >> 2;
    const float4* __restrict__ o4 = (const float4*)obj;
    for (int i = tid; i < n4; i += stride) {
        float4 v = o4[i];
        acc += (softplusf(v.x) + softplusf(v.y)) + (softplusf(v.z) + softplusf(v.w));
    }
    for (int i = (n4 << 2) + tid; i < n; i += stride)
        acc += softplusf(obj[i]);

    // all threads reach here: EXEC full for the WMMA reduction
    float wsum = wave_sum_wmma(acc);

    __shared__ float sh[K1_THREADS / 32];
    const int lane = threadIdx.x & 31;
    const int wave = threadIdx.x >> 5;
    if (lane == 0) sh[wave] = wsum;
    __syncthreads();
    if (threadIdx.x == 0) {
        float s = 0.0f;
        const int nw = blockDim.x >> 5;
        for (int w = 0; w < nw; ++w) s += sh[w];
        partials[blockIdx.x] = s;
    }
}

// ---------------- Kernel 2: assignment + sparse losses + finalize ----------
// One block of B*N threads (512). Winner = first valid box per (batch, cell).
__global__ void __launch_bounds__(1024)
bev_finalize_loss(const float* __restrict__ cls_logits,
                  const float* __restrict__ obj_logits,
                  const float* __restrict__ box_preds,
                  const float* __restrict__ gt_boxes,
                  const int*   __restrict__ gt_labels,
                  const float* __restrict__ gt_masks,
                  const float* __restrict__ partials,
                  float* __restrict__ out,
                  int B, int N, int C, int D, int n_cells_total, int n_partials) {
    const int t  = threadIdx.x;           // t in [0, B*N)
    const int b  = t / N;

    __shared__ int   sh_idx[1024];
    __shared__ float sh_sums[32][8];

    // ---- per-box cell assignment ----
    const float* gtb = gt_boxes + (size_t)t * D;
    const float  x   = gtb[0];
    const float  y   = gtb[1];
    const int    lab = gt_labels[t];
    const float  msk = gt_masks[t];

    const bool inb = (x >= X_MIN) && (x <= X_MAX) && (y >= Y_MIN) && (y <= Y_MAX);
    int gx = (int)floorf((x - X_MIN) / RES);
    int gy = (int)floorf((y - Y_MIN) / RES);
    gx = min(max(gx, 0), BEV_W - 1);
    gy = min(max(gy, 0), BEV_H - 1);
    const int  idx   = gy * BEV_W + gx;
    const bool valid = (msk > 0.5f) && (lab >= 0) && inb;

    sh_idx[t] = valid ? idx : -1;
    __syncthreads();

    // first-come-wins within the same batch
    bool winner = valid;
    if (valid) {
        for (int p = b * N; p < t; ++p) {
            if (sh_idx[p] == idx) { winner = false; break; }
        }
    }

    // ---- per-thread loss contributions (zero for non-winners) ----
    float cnt = 0.0f, sp_w = 0.0f, sp_n = 0.0f, ce = 0.0f, sl1 = 0.0f;
    if (winner) {
        const size_t cell = (size_t)b * NUM_CELLS + (size_t)idx;
        cnt = 1.0f;

        const float xw = obj_logits[cell];
        sp_w = softplusf(xw);
        sp_n = softplusf(-xw);

        // cross-entropy at the winner cell (stable log-softmax)
        const float* cl = cls_logits + cell * (size_t)C;
        float m = cl[0];
        for (int c = 1; c < C; ++c) m = fmaxf(m, cl[c]);
        float s = 0.0f;
        for (int c = 0; c < C; ++c) s += expf(cl[c] - m);
        const int tgt = max(lab, 0);
        ce = (m + logf(s)) - cl[tgt];

        // smooth L1 (beta = 1) vs the GT box
        const float* bp = box_preds + cell * (size_t)D;
        for (int di = 0; di < D; ++di) {
            float d  = bp[di] - gtb[di];
            float ad = fabsf(d);
            sl1 += (ad < 1.0f) ? (0.5f * d * d) : (ad - 0.5f);
        }
    }

    // fold kernel-1 partials (deterministic, fixed order per thread)
    float obj_acc = 0.0f;
    for (int i = t; i < n_partials; i += blockDim.x) obj_acc += partials[i];

    // ---- block reduction: 6 exact WMMA wave sums, then 16 waves via LDS ----
    const int lane = t & 31;
    const int wave = t >> 5;
    float r0 = wave_sum_wmma(cnt);
    float r1 = wave_sum_wmma(sp_w);
    float r2 = wave_sum_wmma(sp_n);
    float r3 = wave_sum_wmma(ce);
    float r4 = wave_sum_wmma(sl1);
    float r5 = wave_sum_wmma(obj_acc);
    if (lane == 0) {
        sh_sums[wave][0] = r0; sh_sums[wave][1] = r1; sh_sums[wave][2] = r2;
        sh_sums[wave][3] = r3; sh_sums[wave][4] = r4; sh_sums[wave][5] = r5;
    }
    __syncthreads();

    if (t == 0) {
        float T[6] = {0, 0, 0, 0, 0, 0};
        const int nw = blockDim.x >> 5;
        for (int w = 0; w < nw; ++w)
            for (int q = 0; q < 6; ++q) T[q] += sh_sums[w][q];

        const float positive    = T[0];
        const float total_cells = (float)n_cells_total;
        const float negatives   = total_cells - positive;
        const float pos_weight  = fmaxf(1.0f, negatives / (positive + 1e-6f));

        // mean over all cells: sum softplus(x) everywhere, swap winner cells'
        // softplus(x) for pos_weight * softplus(-x)
        const float obj_loss = (T[5] - T[1] + pos_weight * T[2]) / total_cells;
        const float cls_loss = (positive > 0.0f) ? (T[3] / fmaxf(positive, 1.0f)) : 0.0f;
        const float box_loss = (positive > 0.0f)
                                   ? (T[4] / fmaxf(positive * (float)D, 1.0f)) : 0.0f;
        const float total = obj_loss + cls_loss + box_loss;  // weights = 1.0

        out[0] = total;
        out[1] = cls_loss;
        out[2] = box_loss;
        out[3] = obj_loss;
    }
}

extern "C" void kernel_launch(void* const* d_in, const int* in_sizes, int n_in,
                              void* d_out, int out_size, void* d_ws, size_t ws_size,
                              hipStream_t stream) {
    const float* cls_logits = (const float*)d_in[0];
    const float* obj_logits = (const float*)d_in[1];
    const float* box_preds  = (const float*)d_in[2];
    const float* gt_boxes   = (const float*)d_in[3];
    const int*   gt_labels  = (const int*)d_in[4];
    const float* gt_masks   = (const float*)d_in[5];

    const int n_obj = in_sizes[1];                 // B * NUM_CELLS
    const int BN    = in_sizes[4];                 // B * N
    const int B     = n_obj / NUM_CELLS;
    const int N     = BN / B;
    const int C     = in_sizes[0] / in_sizes[1];   // 10
    const int D     = in_sizes[2] / in_sizes[1];   // 7

    float* partials = (float*)d_ws;                // K1_BLOCKS floats

    bev_obj_softplus_partials<<<K1_BLOCKS, K1_THREADS, 0, stream>>>(
        obj_logits, partials, n_obj);

    bev_finalize_loss<<<1, BN, 0, stream>>>(
        cls_logits, obj_logits, box_preds, gt_boxes, gt_labels, gt_masks,
        partials, (float*)d_out, B, N, C, D, n_obj, K1_BLOCKS);
}